// RT_28595892257563
// MI455X (gfx1250) — compile-verified
//
#include <hip/hip_runtime.h>
#include <math.h>

typedef __attribute__((ext_vector_type(2))) float v2f;
typedef __attribute__((ext_vector_type(8))) float v8f;

#define Hh 64
#define FDIM 128
#define LDSS 132   // 128 + 4 pad: wave-wide v2f reads hit all 64 banks exactly once

__device__ __forceinline__ float waveReduce(float v) {
    #pragma unroll
    for (int off = 16; off >= 1; off >>= 1) v += __shfl_xor(v, off, 32);
    return v;
}

// ---------------------------------------------------------------------------
// h = elu(feat @ W.T + b)  via V_WMMA_F32_16X16X4_F32 (exact fp32 matmul)
// block = 128 threads (4 waves), 64 rows / block, wave owns 16 rows x 64 cols
// ---------------------------------------------------------------------------
__global__ __launch_bounds__(128)
void proj_kernel(const float* __restrict__ feat, const float* __restrict__ W,
                 const float* __restrict__ bias, float* __restrict__ h, int N)
{
    __shared__ float sA[64 * LDSS];
    const int tid = threadIdx.x;
    const int rowBase = blockIdx.x * 64;

    // cooperative, coalesced stage of the 64x128 feature tile
    for (int f = tid; f < 64 * (FDIM / 4); f += 128) {
        int r  = f >> 5;        // 32 float4 per row
        int c4 = f & 31;
        float4 v = make_float4(0.f, 0.f, 0.f, 0.f);
        int row = rowBase + r;
        if (row < N) v = *(const float4*)(feat + (size_t)row * FDIM + c4 * 4);
        float* dst = &sA[r * LDSS + c4 * 4];
        v2f lo; lo.x = v.x; lo.y = v.y;
        v2f hi; hi.x = v.z; hi.y = v.w;
        *(v2f*)(dst)     = lo;
        *(v2f*)(dst + 2) = hi;
    }
    __syncthreads();

    const int wave = tid >> 5;
    const int lane = tid & 31;
    const int m    = lane & 15;
    const int kk   = (lane >> 4) * 2;   // A/B fragment K sub-offset per ISA layout
    const float* aBase = &sA[(wave * 16 + m) * LDSS];

    v8f c0 = {}, c1 = {}, c2 = {}, c3 = {};
    #pragma unroll 4
    for (int k0 = 0; k0 < FDIM; k0 += 4) {
        v2f a = *(const v2f*)(aBase + k0 + kk);
        const float* wB = W + (size_t)(k0 + kk);   // B[k][n] = W[n][k], hot in WGP$
        v2f bv0 = *(const v2f*)(wB + (size_t)( 0 + m) * FDIM);
        v2f bv1 = *(const v2f*)(wB + (size_t)(16 + m) * FDIM);
        v2f bv2 = *(const v2f*)(wB + (size_t)(32 + m) * FDIM);
        v2f bv3 = *(const v2f*)(wB + (size_t)(48 + m) * FDIM);
        c0 = __builtin_amdgcn_wmma_f32_16x16x4_f32(false, a, false, bv0, (short)0, c0, false, false);
        c1 = __builtin_amdgcn_wmma_f32_16x16x4_f32(false, a, false, bv1, (short)0, c1, false, false);
        c2 = __builtin_amdgcn_wmma_f32_16x16x4_f32(false, a, false, bv2, (short)0, c2, false, false);
        c3 = __builtin_amdgcn_wmma_f32_16x16x4_f32(false, a, false, bv3, (short)0, c3, false, false);
    }

    // D layout: VGPR r -> (M=r, N=lane) for lanes 0-15, (M=r+8, N=lane-16) for 16-31
    const int rOff = rowBase + wave * 16 + ((lane >> 4) ? 8 : 0);
    v8f cc[4] = {c0, c1, c2, c3};
    #pragma unroll
    for (int nt = 0; nt < 4; ++nt) {
        int n = nt * 16 + m;
        float bv = bias[n];
        #pragma unroll
        for (int r = 0; r < 8; ++r) {
            int row = rOff + r;
            if (row < N) {
                float x = cc[nt][r] + bv;
                h[(size_t)row * Hh + n] = (x > 0.f) ? x : expm1f(x);   // elu
            }
        }
    }
}

// ---------------------------------------------------------------------------
// Node-level attention: one wave per target node, online softmax over S nbrs
// ---------------------------------------------------------------------------
__global__ __launch_bounds__(256)
void node_att_kernel(const float* __restrict__ target_h, const float* __restrict__ nei_h,
                     const int* __restrict__ nei, const float* __restrict__ att,
                     float* __restrict__ out, int N0, int S)
{
    const int wave = threadIdx.x >> 5;
    const int lane = threadIdx.x & 31;
    const int node = blockIdx.x * 8 + wave;
    if (node >= N0) return;

    v2f atv = *(const v2f*)(att + 2 * lane);          // a_t
    v2f anv = *(const v2f*)(att + Hh + 2 * lane);     // a_n
    v2f hv  = *(const v2f*)(target_h + (size_t)node * Hh + 2 * lane);
    float t = waveReduce(hv.x * atv.x + hv.y * atv.y);

    float mx = -INFINITY, den = 0.f, accx = 0.f, accy = 0.f;
    const int* nrow = nei + (size_t)node * S;
    for (int s = 0; s < S; ++s) {
        int j = nrow[s];
        v2f ev = *(const v2f*)(nei_h + (size_t)j * Hh + 2 * lane);  // L2-resident gather
        float e = waveReduce(ev.x * anv.x + ev.y * anv.y);
        float x = t + e;
        x = (x > 0.f) ? x : 0.01f * x;                 // leaky_relu(0.01)
        float nm = fmaxf(mx, x);
        float sc = expf(mx - nm);                      // first iter: exp(-inf)=0
        float w  = expf(x - nm);
        den  = den  * sc + w;
        accx = accx * sc + w * ev.x;
        accy = accy * sc + w * ev.y;
        mx = nm;
    }
    float rx = accx / den, ry = accy / den;
    v2f o;
    o.x = (rx > 0.f) ? rx : expm1f(rx);                // elu
    o.y = (ry > 0.f) ? ry : expm1f(ry);
    *(v2f*)(out + (size_t)node * Hh + 2 * lane) = o;
}

// ---------------------------------------------------------------------------
// Type-level attention partial: accum += sum_n tanh(e_n @ fc_w.T + fc_b) @ att_sem
// ---------------------------------------------------------------------------
__global__ __launch_bounds__(256)
void type_att_partial(const float* __restrict__ emb, const float* __restrict__ fc_w,
                      const float* __restrict__ fc_b, const float* __restrict__ att_sem,
                      float* __restrict__ accum, int N0)
{
    __shared__ float sh[8][Hh];
    const int wave = threadIdx.x >> 5;
    const int lane = threadIdx.x & 31;
    const int node = blockIdx.x * 8 + wave;
    if (node >= N0) return;

    v2f ev = *(const v2f*)(emb + (size_t)node * Hh + 2 * lane);
    sh[wave][2 * lane]     = ev.x;
    sh[wave][2 * lane + 1] = ev.y;
    // wave-private LDS exchange: DS pipeline is in-order per wave, no block barrier
    asm volatile("s_wait_dscnt 0" ::: "memory");

    float part = 0.f;
    #pragma unroll
    for (int jj = 0; jj < 2; ++jj) {
        int j = lane + jj * 32;
        const float* wrow = fc_w + (size_t)j * Hh;     // 16 KB, hot in cache
        float s = fc_b[j];
        #pragma unroll
        for (int k = 0; k < Hh; k += 4) {
            float4 wv = *(const float4*)(wrow + k);
            float4 xv = *(const float4*)(&sh[wave][k]); // broadcast read
            s += wv.x * xv.x + wv.y * xv.y + wv.z * xv.z + wv.w * xv.w;
        }
        part += tanhf(s) * att_sem[j];
    }
    float tot = waveReduce(part);
    if (lane == 0) atomicAdd(accum, tot);
}

__global__ void zero2_kernel(float* p) { p[0] = 0.f; p[1] = 0.f; }

// ---------------------------------------------------------------------------
// z = softmax([beta_t, beta_s])[0]*v_teacher + [1]*v_student
// ---------------------------------------------------------------------------
__global__ __launch_bounds__(256)
void combine_kernel(const float* __restrict__ vt, const float* __restrict__ vs,
                    const float* __restrict__ betaAcc, float* __restrict__ out,
                    int total, float invN)
{
    float b0 = betaAcc[0] * invN;                      // teacher
    float b1 = betaAcc[1] * invN;                      // student
    float mm = fmaxf(b0, b1);
    float e0 = expf(b0 - mm), e1 = expf(b1 - mm);
    float w0 = e0 / (e0 + e1), w1 = e1 / (e0 + e1);
    for (int i = blockIdx.x * blockDim.x + threadIdx.x; i < total;
         i += gridDim.x * blockDim.x)
        out[i] = w0 * vt[i] + w1 * vs[i];
}

extern "C" void kernel_launch(void* const* d_in, const int* in_sizes, int n_in,
                              void* d_out, int out_size, void* d_ws, size_t ws_size,
                              hipStream_t stream)
{
    const float* feat0 = (const float*)d_in[0];
    const float* feat1 = (const float*)d_in[1];
    const float* feat2 = (const float*)d_in[2];
    const float* W0 = (const float*)d_in[3];
    const float* b0 = (const float*)d_in[4];
    const float* W1 = (const float*)d_in[5];
    const float* b1 = (const float*)d_in[6];
    const float* W2 = (const float*)d_in[7];
    const float* b2 = (const float*)d_in[8];
    const float* att0    = (const float*)d_in[9];
    const float* att1    = (const float*)d_in[10];
    const float* fc_w    = (const float*)d_in[11];
    const float* fc_b    = (const float*)d_in[12];
    const float* att_sem = (const float*)d_in[13];
    const int* nei0 = (const int*)d_in[14];
    const int* nei1 = (const int*)d_in[15];

    const int N0 = in_sizes[0] / FDIM;
    const int N1 = in_sizes[1] / FDIM;
    const int N2 = in_sizes[2] / FDIM;
    const int S0 = in_sizes[14] / N0;
    const int S1 = in_sizes[15] / N0;

    float* ws = (float*)d_ws;
    size_t o = 0;
    float* h0   = ws + o; o += (size_t)N0 * Hh;
    float* h1   = ws + o; o += (size_t)N1 * Hh;
    float* h2   = ws + o; o += (size_t)N2 * Hh;
    float* vstu = ws + o; o += (size_t)N0 * Hh;
    float* vtea = ws + o; o += (size_t)N0 * Hh;
    float* betaAcc = ws + o;

    proj_kernel<<<(N0 + 63) / 64, 128, 0, stream>>>(feat0, W0, b0, h0, N0);
    proj_kernel<<<(N1 + 63) / 64, 128, 0, stream>>>(feat1, W1, b1, h1, N1);
    proj_kernel<<<(N2 + 63) / 64, 128, 0, stream>>>(feat2, W2, b2, h2, N2);

    node_att_kernel<<<(N0 + 7) / 8, 256, 0, stream>>>(h0, h1, nei0, att0, vstu, N0, S0);
    node_att_kernel<<<(N0 + 7) / 8, 256, 0, stream>>>(h0, h2, nei1, att1, vtea, N0, S1);

    zero2_kernel<<<1, 1, 0, stream>>>(betaAcc);
    type_att_partial<<<(N0 + 7) / 8, 256, 0, stream>>>(vtea, fc_w, fc_b, att_sem, betaAcc + 0, N0);
    type_att_partial<<<(N0 + 7) / 8, 256, 0, stream>>>(vstu, fc_w, fc_b, att_sem, betaAcc + 1, N0);

    int total = N0 * Hh;
    combine_kernel<<<(total + 255) / 256, 256, 0, stream>>>(vtea, vstu, betaAcc, (float*)d_out,
                                                            total, 1.0f / (float)N0);
}